// SpRelWeiADJ_64639257805012
// MI455X (gfx1250) — compile-verified
//
#include <hip/hip_runtime.h>
#include <hip/hip_bf16.h>
#include <math.h>

typedef float v2f __attribute__((ext_vector_type(2)));
typedef float v8f __attribute__((ext_vector_type(8)));

#define DIM     256
#define NSR     1000
#define NTG     1200
#define EPSF    1e-8f
#define ROW_TILES ((NSR + 15) / 16)   // 63 (covers rows 0..1007; tail rows are don't-care)
#define COL_TILES (NTG / 16)          // 75

// ---------------------------------------------------------------------------
// Init the two attention tables. att_a (row max over sr rows) starts at -2
// (cosine sims are >= -1). att_b (col max) starts at 0: the reference pads sr
// with zero rows whose normalized dot products are exactly 0, so every column
// max implicitly includes 0.
// ---------------------------------------------------------------------------
__global__ void init_att_kernel(float* __restrict__ att_a, float* __restrict__ att_b) {
  int i = blockIdx.x * blockDim.x + threadIdx.x;
  if (i < 1216) {
    att_a[i] = -2.0f;
    att_b[i] = 0.0f;
  }
}

// ---------------------------------------------------------------------------
// Per-row 1/(||x|| + eps). One wave32 per row; lane-strided partial sums,
// butterfly reduce with __shfl_xor.
// rows [0, NSR)          -> rel_embedding_sr -> inv_a
// rows [NSR, NSR+NTG)    -> rel_embedding_tg -> inv_b
// ---------------------------------------------------------------------------
__global__ void row_inv_norm_kernel(const float* __restrict__ A,
                                    const float* __restrict__ B,
                                    float* __restrict__ inv_a,
                                    float* __restrict__ inv_b) {
  int wave = (blockIdx.x * blockDim.x + threadIdx.x) >> 5;
  int lane = threadIdx.x & 31;
  if (wave >= NSR + NTG) return;

  const float* src;
  float* dst;
  if (wave < NSR) { src = A + wave * DIM;         dst = inv_a + wave; }
  else            { src = B + (wave - NSR) * DIM; dst = inv_b + (wave - NSR); }

  float s = 0.0f;
  #pragma unroll
  for (int k = 0; k < DIM / 32; ++k) {
    float v = src[lane + 32 * k];
    s += v * v;
  }
  #pragma unroll
  for (int m = 16; m >= 1; m >>= 1) s += __shfl_xor(s, m, 32);

  if (lane == 0) *dst = 1.0f / (sqrtf(s) + EPSF);
}

// ---------------------------------------------------------------------------
// One wave per 16x16 sim tile. K=256 accumulated as 64 chained
// v_wmma_f32_16x16x4_f32 ops, 4 K-steps per loop iteration with all 8 operand
// loads issued before the WMMA group (loads in flight over the matrix pipe).
//
// Tail A rows (1000..1007) are loaded from a clamped in-bounds row; their
// results are forced to 0 in the epilogue via ia = 0, which matches the
// reference's zero-padded rows. This keeps the loop fully branchless (no
// per-load EXEC save/restore).
//
// ISA operand layouts (wave32):
//   A 16x4 f32:  VGPR0: lanes0-15 K=0, lanes16-31 K=2; VGPR1: K=1 / K=3; M=lane%16
//   B  4x16 f32: mirrored with N=lane%16, same K split across halves
//   -> both operands load float2 at base + (lane%16)*DIM + 2*(lane>>4) + k0
//   C/D 16x16 f32: VGPR v: lanes0-15 -> M=v, lanes16-31 -> M=v+8; N=lane%16
// ---------------------------------------------------------------------------
__global__ void sim_max_wmma_kernel(const float* __restrict__ A,
                                    const float* __restrict__ B,
                                    const float* __restrict__ inv_a,
                                    const float* __restrict__ inv_b,
                                    float* __restrict__ att_a,
                                    float* __restrict__ att_b) {
  int wave = (blockIdx.x * blockDim.x + threadIdx.x) >> 5;
  if (wave >= ROW_TILES * COL_TILES) return;   // wave-uniform exit; EXEC all-1s for WMMA

  int lane = threadIdx.x & 31;
  int lm = lane & 15;    // M (for A) / N (for B,C,D)
  int lh = lane >> 4;    // lane half selects K pair

  int rt = wave / COL_TILES;
  int ct = wave % COL_TILES;
  int row0 = rt * 16;
  int col0 = ct * 16;

  int arow = row0 + lm;
  int arow_c = arow < NSR ? arow : (NSR - 1);  // clamp: value is don't-care, zeroed in epilogue
  int brow = col0 + lm;
  const float* aptr = A + arow_c * DIM + 2 * lh;
  const float* bptr = B + brow * DIM + 2 * lh;

  v8f c = {0.f, 0.f, 0.f, 0.f, 0.f, 0.f, 0.f, 0.f};
  for (int k0 = 0; k0 < DIM; k0 += 16) {
    // 8 branchless b64 loads up front, then 4 dependent WMMAs
    v2f a0 = *(const v2f*)(aptr + k0);
    v2f b0 = *(const v2f*)(bptr + k0);
    v2f a1 = *(const v2f*)(aptr + k0 + 4);
    v2f b1 = *(const v2f*)(bptr + k0 + 4);
    v2f a2 = *(const v2f*)(aptr + k0 + 8);
    v2f b2 = *(const v2f*)(bptr + k0 + 8);
    v2f a3 = *(const v2f*)(aptr + k0 + 12);
    v2f b3 = *(const v2f*)(bptr + k0 + 12);
    c = __builtin_amdgcn_wmma_f32_16x16x4_f32(false, a0, false, b0, (short)0, c, false, false);
    c = __builtin_amdgcn_wmma_f32_16x16x4_f32(false, a1, false, b1, (short)0, c, false, false);
    c = __builtin_amdgcn_wmma_f32_16x16x4_f32(false, a2, false, b2, (short)0, c, false, false);
    c = __builtin_amdgcn_wmma_f32_16x16x4_f32(false, a3, false, b3, (short)0, c, false, false);
  }

  // ---- epilogue: scale + fused row/col max ----
  float ib = inv_b[col0 + lm];   // every element this lane holds has N = lm
  float colmax = 0.0f;           // padded rows contribute 0 to column maxes

  #pragma unroll
  for (int v = 0; v < 8; ++v) {
    int m = row0 + v + 8 * lh;   // row of element v in this lane
    float ia = (m < NSR) ? inv_a[m] : 0.0f;   // zeroes the clamped tail rows
    float s = c[v] * ia * ib;
    colmax = fmaxf(colmax, s);

    // row max: reduce over the 16 lanes of this half (xor masks keep bit4)
    float rmax = s;
    #pragma unroll
    for (int msk = 8; msk >= 1; msk >>= 1) rmax = fmaxf(rmax, __shfl_xor(rmax, msk, 32));
    if (lm == 0) atomicMax(&att_a[m], rmax);   // lanes 0 and 16 cover rows v and v+8
  }

  // col max: combine the two halves (same N per lm), one atomic per column
  colmax = fmaxf(colmax, __shfl_xor(colmax, 16, 32));
  if (lh == 0) atomicMax(&att_b[col0 + lm], colmax);
}

// ---------------------------------------------------------------------------
// Streaming gather, 4 edges/thread with b128 loads/stores:
//   out[i]      = min(att_a[rel_sr[i]], 1)
//   out[NE + i] = min(att_b[rel_tg[i]], 1)
// Tables are ~4.8 KB -> cache resident; this is the HBM-bound phase
// (~32 MB total traffic => ~1.4 us at 23.3 TB/s).
// ---------------------------------------------------------------------------
__global__ void gather4_kernel(const int4* __restrict__ rel_sr4,
                               const int4* __restrict__ rel_tg4,
                               const float* __restrict__ att_a,
                               const float* __restrict__ att_b,
                               float* __restrict__ out,
                               int n4, int n_edge) {
  int i = blockIdx.x * blockDim.x + threadIdx.x;
  if (i < n4) {
    int4 s = rel_sr4[i];
    int4 t = rel_tg4[i];
    float4 os;
    os.x = fminf(att_a[s.x], 1.0f);
    os.y = fminf(att_a[s.y], 1.0f);
    os.z = fminf(att_a[s.z], 1.0f);
    os.w = fminf(att_a[s.w], 1.0f);
    float4 ot;
    ot.x = fminf(att_b[t.x], 1.0f);
    ot.y = fminf(att_b[t.y], 1.0f);
    ot.z = fminf(att_b[t.z], 1.0f);
    ot.w = fminf(att_b[t.w], 1.0f);
    ((float4*)out)[i] = os;
    ((float4*)(out + n_edge))[i] = ot;
  }
}

// scalar tail for n_edge % 4 != 0 (not hit for N_EDGE = 2M, kept for generality)
__global__ void gather_tail_kernel(const int* __restrict__ rel_sr,
                                   const int* __restrict__ rel_tg,
                                   const float* __restrict__ att_a,
                                   const float* __restrict__ att_b,
                                   float* __restrict__ out,
                                   int start, int n_edge) {
  int i = start + blockIdx.x * blockDim.x + threadIdx.x;
  if (i < n_edge) {
    out[i]          = fminf(att_a[rel_sr[i]], 1.0f);
    out[n_edge + i] = fminf(att_b[rel_tg[i]], 1.0f);
  }
}

extern "C" void kernel_launch(void* const* d_in, const int* in_sizes, int n_in,
                              void* d_out, int out_size, void* d_ws, size_t ws_size,
                              hipStream_t stream) {
  const float* A      = (const float*)d_in[0];  // rel_embedding_sr (1000x256)
  const float* B      = (const float*)d_in[1];  // rel_embedding_tg (1200x256)
  const int*   rel_sr = (const int*)d_in[2];    // (N_EDGE)
  const int*   rel_tg = (const int*)d_in[3];    // (N_EDGE)
  // d_in[4], d_in[5] (pos_sr, pos_tg) are unused by the reference
  float* out = (float*)d_out;
  int n_edge = in_sizes[2];

  // workspace layout (floats): att_a[1216] | att_b[1216] | inv_a[1024] | inv_b[1216]
  float* att_a = (float*)d_ws;
  float* att_b = att_a + 1216;
  float* inv_a = att_b + 1216;
  float* inv_b = inv_a + 1024;

  init_att_kernel<<<(1216 + 255) / 256, 256, 0, stream>>>(att_a, att_b);

  // one wave32 per row; 256-thread blocks = 8 waves/block
  row_inv_norm_kernel<<<(NSR + NTG + 7) / 8, 256, 0, stream>>>(A, B, inv_a, inv_b);

  int tiles = ROW_TILES * COL_TILES;            // 4725 waves
  sim_max_wmma_kernel<<<(tiles + 7) / 8, 256, 0, stream>>>(A, B, inv_a, inv_b, att_a, att_b);

  int n4 = n_edge >> 2;
  if (n4 > 0) {
    gather4_kernel<<<(n4 + 255) / 256, 256, 0, stream>>>(
        (const int4*)rel_sr, (const int4*)rel_tg, att_a, att_b, out, n4, n_edge);
  }
  int tail = n_edge - (n4 << 2);
  if (tail > 0) {
    gather_tail_kernel<<<(tail + 255) / 256, 256, 0, stream>>>(
        rel_sr, rel_tg, att_a, att_b, out, n4 << 2, n_edge);
  }
}